// FocalLossAdaptive_65566970741113
// MI455X (gfx1250) — compile-verified
//
#include <hip/hip_runtime.h>
#include <math.h>

// Focal loss with adaptive gamma, N=8192 rows, C=32000 classes, fp32.
// Memory-bound: 1.05 GB single-pass stream -> ~45us floor at 23.3 TB/s.

#define BLK 256

typedef __attribute__((ext_vector_type(4))) float v4f;
typedef __attribute__((ext_vector_type(2))) float v2f;
typedef __attribute__((ext_vector_type(8))) float v8f;

#define NEG_FLT_MAX (-3.402823466e38f)

// Merge two online-softmax partial states: (m,s) <- combine((m,s),(m2,s2))
__device__ __forceinline__ void online_merge(float& m, float& s, float m2, float s2) {
    float M = fmaxf(m, m2);
    s = s * __expf(m - M) + s2 * __expf(m2 - M);
    m = M;
}

__global__ __launch_bounds__(BLK) void focal_rows_kernel(
    const float* __restrict__ in,   // [N, C]
    const int*   __restrict__ tgt,  // [N]
    float*       __restrict__ row_loss, // [N] workspace
    int C)
{
    __shared__ float sm[BLK];
    __shared__ float ss[BLK];

    const int row = blockIdx.x;
    const int tid = threadIdx.x;
    const int nv4 = C >> 2;                       // C % 4 == 0 (32000)
    const v4f* rp = (const v4f*)(in + (size_t)row * (size_t)C);

    // 4 independent online-softmax accumulators for ILP.
    float m0 = NEG_FLT_MAX, m1 = NEG_FLT_MAX, m2 = NEG_FLT_MAX, m3 = NEG_FLT_MAX;
    float s0 = 0.f, s1 = 0.f, s2 = 0.f, s3 = 0.f;

    for (int j = tid; j < nv4; j += BLK) {
        // Keep the stream deep: prefetch 4 block-strides (16 KB) ahead.
        if (j + 4 * BLK < nv4) __builtin_prefetch((const void*)(rp + j + 4 * BLK), 0, 0);
        // Data is touched exactly once: non-temporal load, don't pollute caches.
        v4f v = __builtin_nontemporal_load(rp + j);
        float nm;
        nm = fmaxf(m0, v.x); s0 = s0 * __expf(m0 - nm) + __expf(v.x - nm); m0 = nm;
        nm = fmaxf(m1, v.y); s1 = s1 * __expf(m1 - nm) + __expf(v.y - nm); m1 = nm;
        nm = fmaxf(m2, v.z); s2 = s2 * __expf(m2 - nm) + __expf(v.z - nm); m2 = nm;
        nm = fmaxf(m3, v.w); s3 = s3 * __expf(m3 - nm) + __expf(v.w - nm); m3 = nm;
    }

    online_merge(m0, s0, m1, s1);
    online_merge(m2, s2, m3, s3);
    online_merge(m0, s0, m2, s2);

    sm[tid] = m0;
    ss[tid] = s0;
    __syncthreads();

    // Block tree-reduce of the online-softmax state.
    for (int off = BLK / 2; off > 0; off >>= 1) {
        if (tid < off) {
            float m = sm[tid], s = ss[tid];
            online_merge(m, s, sm[tid + off], ss[tid + off]);
            sm[tid] = m;
            ss[tid] = s;
        }
        __syncthreads();
    }

    if (tid == 0) {
        const float M   = sm[0];
        const float S   = ss[0];
        const float lse = M + __logf(S);
        const float xt  = in[(size_t)row * (size_t)C + (size_t)tgt[row]];
        const float logpt = xt - lse;
        const float pt    = __expf(logpt);
        const float u     = 1.0f - pt;
        const float u3    = u * u * u;
        // gamma: 0 if pt>=0.5 -> w=1 ; 5 if pt<0.2 -> u^5 ; else 3 -> u^3
        const float w = (pt >= 0.5f) ? 1.0f : ((pt < 0.2f) ? u3 * u * u : u3);
        row_loss[row] = -w * logpt;
    }
}

// Deterministic mean over N row losses. Final 32->1 cross-lane sum done with
// V_WMMA_F32_16X16X4_F32 against a ones B-matrix:
//   A[m][0] = v_m, A[m][2] = v_{m+16} (a.x = lane value, a.y = 0)
//   D[m][n] = v_m + v_{m+16}; sum of D VGPRs 0..7 per lane + (lane0 + lane16)
//   equals the full 32-lane sum.
__global__ __launch_bounds__(256) void reduce_mean_kernel(
    const float* __restrict__ row_loss, float* __restrict__ out, int N)
{
    __shared__ float buf[256];
    const int tid = threadIdx.x;

    float acc = 0.f;
    for (int i = tid; i < N; i += 256) acc += row_loss[i];
    buf[tid] = acc;
    __syncthreads();

    if (tid >= 32) return;   // wave 0 continues with EXEC all ones

    float v = 0.f;
#pragma unroll
    for (int j = 0; j < 8; ++j) v += buf[tid + 32 * j];

    v2f a; a.x = v;    a.y = 0.0f;
    v2f b; b.x = 1.0f; b.y = 1.0f;
    v8f c = {};
    c = __builtin_amdgcn_wmma_f32_16x16x4_f32(
            /*neg_a=*/false, a, /*neg_b=*/false, b,
            /*c_mod=*/(short)0, c, /*reuse_a=*/false, /*reuse_b=*/false);

    float t = c[0] + c[1] + c[2] + c[3] + c[4] + c[5] + c[6] + c[7];
    const float t0  = __uint_as_float(__builtin_amdgcn_readlane(__float_as_uint(t), 0));
    const float t16 = __uint_as_float(__builtin_amdgcn_readlane(__float_as_uint(t), 16));

    if (tid == 0) out[0] = (t0 + t16) / (float)N;
}

extern "C" void kernel_launch(void* const* d_in, const int* in_sizes, int n_in,
                              void* d_out, int out_size, void* d_ws, size_t ws_size,
                              hipStream_t stream)
{
    const float* in  = (const float*)d_in[0];
    const int*   tgt = (const int*)d_in[1];

    const int N = in_sizes[1];               // 8192
    const int C = in_sizes[0] / N;           // 32000

    float* row_loss = (float*)d_ws;          // N floats of scratch

    focal_rows_kernel<<<N, BLK, 0, stream>>>(in, tgt, row_loss, C);
    reduce_mean_kernel<<<1, 256, 0, stream>>>(row_loss, (float*)d_out, N);
}